// PointCNN_3272765079552
// MI455X (gfx1250) — compile-verified
//
#include <hip/hip_runtime.h>
#include <hip/hip_bf16.h>

// ---------------------------------------------------------------------------
// PointCNN on MI455X (gfx1250). All GEMM-class work runs on the fp32 WMMA
// pipe (v_wmma_f32_16x16x4_f32) to keep full fp32 precision (K up to 18432).
// GEMM stages tiles through LDS: coalesced b128 global loads, ds_load_b64
// fragment reads, branch-free WMMA inner loop.
// ---------------------------------------------------------------------------

typedef float v2f __attribute__((ext_vector_type(2)));
typedef float v8f __attribute__((ext_vector_type(8)));

#define BN_EPS 1e-5f
#define SELU_ALPHA 1.6732632423543772f
#define SELU_SCALE 1.0507009873554805f

// ---------------------------------------------------------------------------
// GEMM: C[M,N] = A[M,K] * W[K,N] + bias[N]
// Block: 256 threads (8 waves) -> 64(M) x 32(N) tile, K-chunks of 32.
// LDS: As[64][34] row-major (M x K), Bs[32][34] transposed (N x K); the pad
// of 34 keeps each lane's 2-element fragment 8-byte aligned (ds_load_b64)
// and spreads banks. Fragment layout for v_wmma_f32_16x16x4_f32:
//   A: lanes 0-15 row M=l hold K=kk,kk+1 ; lanes 16-31 hold K=kk+2,kk+3.
//   B: mirrored (row=K, col=N).  C/D: VGPR r -> (M=r / r+8, N=l).
// ---------------------------------------------------------------------------
#define GEMM_BK 32

__global__ __launch_bounds__(256) void wmma_gemm_f32(
    const float* __restrict__ A, const float* __restrict__ W,
    const float* __restrict__ bias, float* __restrict__ C,
    int M, int N, int K)
{
  __shared__ float As[64][34];
  __shared__ float Bs[32][34];
  const int tid = threadIdx.x;
  const int wave = tid >> 5, lane = tid & 31;
  const int half = lane >> 4, l = lane & 15;
  const int blockM = blockIdx.y * 64, blockN = blockIdx.x * 32;
  const int wm = (wave & 3) * 16;      // M offset of this wave's tile
  const int wn = (wave >> 2) * 16;     // N offset of this wave's tile
  const int koff = half * 2;

  // staging assignments (all 256 threads participate)
  const int ar = tid >> 2;             // 0..63 : A row within tile
  const int ak = (tid & 3) * 8;        // 0,8,16,24 : A k-start (8 consecutive)
  int arow = blockM + ar; if (arow >= M) arow = M - 1;   // clamped, guarded later
  const int bk = tid >> 3;             // 0..31 : B k within chunk
  const int bc = (tid & 7) * 4;        // 0..28 : B col start (4 consecutive)
  const bool nFull = (blockN + 32 <= N);   // block-uniform: clamp-free W loads

  v8f acc = {};
  for (int kb = 0; kb < K; kb += GEMM_BK) {
    __syncthreads();
    const float* ap = A + (size_t)arow * K + kb + ak;
    if (kb + GEMM_BK <= K) {           // full chunk: branch-free staging
#pragma unroll
      for (int j = 0; j < 8; ++j) As[ar][ak + j] = ap[j];
      const float* wp = W + (size_t)(kb + bk) * N;
      if (nFull) {                     // contiguous -> merges to b128
        const float* wpc = wp + blockN + bc;
#pragma unroll
        for (int j = 0; j < 4; ++j) Bs[bc + j][bk] = wpc[j];
      } else {
#pragma unroll
        for (int j = 0; j < 4; ++j) {
          int c = blockN + bc + j; if (c >= N) c = N - 1;
          Bs[bc + j][bk] = wp[c];
        }
      }
      if (kb + GEMM_BK < K) {
        __builtin_prefetch(ap + GEMM_BK, 0, 0);
        __builtin_prefetch(wp + (size_t)GEMM_BK * N, 0, 0);
      }
    } else {                           // K tail: zero-fill (value selects)
#pragma unroll
      for (int j = 0; j < 8; ++j)
        As[ar][ak + j] = (kb + ak + j < K) ? ap[j] : 0.0f;
      const int gk = kb + bk;
      const float* wp = W + (size_t)(gk < K ? gk : K - 1) * N;
#pragma unroll
      for (int j = 0; j < 4; ++j) {
        int c = blockN + bc + j; if (c >= N) c = N - 1;
        Bs[bc + j][bk] = (gk < K) ? wp[c] : 0.0f;
      }
    }
    __syncthreads();
#pragma unroll
    for (int kt = 0; kt < GEMM_BK; kt += 4) {
      const v2f a = *(const v2f*)&As[wm + l][kt + koff];
      const v2f b = *(const v2f*)&Bs[wn + l][kt + koff];
      acc = __builtin_amdgcn_wmma_f32_16x16x4_f32(false, a, false, b,
                                                  (short)0, acc, false, false);
    }
  }

  const int col = blockN + wn + l;
  const int colc = (col < N) ? col : N - 1;
  const float bv = bias ? bias[colc] : 0.0f;
#pragma unroll
  for (int rr = 0; rr < 8; ++rr) {
    const int row = blockM + wm + rr + 8 * half;
    if (row < M && col < N) C[(size_t)row * N + col] = acc[rr] + bv;
  }
}

// ---------------------------------------------------------------------------
// Per-(b,s) block-diagonal matmul: G[bs] = X[bs](k x k) @ F[bs](k x C)
// One wave per (bs, C-tile). k is 8/12/16 here (multiple of 4) -> no K guards.
// ---------------------------------------------------------------------------
__global__ __launch_bounds__(256) void wmma_bmm_xfeats(
    const float* __restrict__ Xm,   // [BS, k*k]
    const float* __restrict__ F,    // [BS, k*C]
    float* __restrict__ G,          // [BS, k*C]
    int BS, int k, int C)
{
  const int ctiles = (C + 15) >> 4;
  const int gw = blockIdx.x * 8 + (threadIdx.x >> 5);
  if (gw >= BS * ctiles) return;    // uniform per wave
  const int bs = gw / ctiles, ct = gw % ctiles;
  const int lane = threadIdx.x & 31;
  const int half = lane >> 4;
  const int l = lane & 15;
  const float* Xp = Xm + (size_t)bs * k * k;
  const float* Fp = F + (size_t)bs * k * C;
  float* Gp = G + (size_t)bs * k * C;

  const int r = (l < k) ? l : (k - 1);            // clamped row, stores guarded
  int c = ct * 16 + l; if (c >= C) c = C - 1;     // clamped col, stores guarded
  const int koff = half * 2;

  v8f acc = {};
  for (int kk = 0; kk < k; kk += 4) {
    const int k0 = kk + koff;
    v2f a, b;
    a.x = Xp[r * k + k0];
    a.y = Xp[r * k + k0 + 1];
    b.x = Fp[(size_t)k0 * C + c];
    b.y = Fp[(size_t)(k0 + 1) * C + c];
    acc = __builtin_amdgcn_wmma_f32_16x16x4_f32(false, a, false, b,
                                                (short)0, acc, false, false);
  }
#pragma unroll
  for (int rr = 0; rr < 8; ++rr) {
    const int row = rr + 8 * half;
    const int col = ct * 16 + l;
    if (row < k && col < C) Gp[(size_t)row * C + col] = acc[rr];
  }
}

// ---------------------------------------------------------------------------
// Furthest point sampling: one block per batch, dist[] resident in LDS.
// Matches JAX: idx[0]=0, then S-1 argmax steps (ties -> smallest index).
// ---------------------------------------------------------------------------
__global__ __launch_bounds__(512) void fps_kernel(
    const float* __restrict__ P, int Np, int S, int* __restrict__ idxOut)
{
  extern __shared__ float dist[];             // Np floats
  __shared__ float rv[512];
  __shared__ int ri[512];
  __shared__ float lastp[3];
  const int b = blockIdx.x;
  const float* Pb = P + (size_t)b * Np * 3;
  for (int i = threadIdx.x; i < Np; i += 512) dist[i] = 1e10f;
  if (threadIdx.x == 0) {
    idxOut[(size_t)b * S] = 0;
    lastp[0] = Pb[0]; lastp[1] = Pb[1]; lastp[2] = Pb[2];
  }
  __syncthreads();
  for (int j = 1; j < S; ++j) {
    const float lx = lastp[0], ly = lastp[1], lz = lastp[2];
    float best = -1.0f; int bi = 0x7fffffff;
    for (int i = threadIdx.x; i < Np; i += 512) {
      const float dx = Pb[i*3+0]-lx, dy = Pb[i*3+1]-ly, dz = Pb[i*3+2]-lz;
      const float d = dx*dx + dy*dy + dz*dz;
      const float nd = fminf(dist[i], d);
      dist[i] = nd;
      if (nd > best || (nd == best && i < bi)) { best = nd; bi = i; }
    }
    rv[threadIdx.x] = best; ri[threadIdx.x] = bi;
    __syncthreads();
    for (int st = 256; st > 0; st >>= 1) {
      if (threadIdx.x < st) {
        const float ov = rv[threadIdx.x + st]; const int oi = ri[threadIdx.x + st];
        if (ov > rv[threadIdx.x] || (ov == rv[threadIdx.x] && oi < ri[threadIdx.x])) {
          rv[threadIdx.x] = ov; ri[threadIdx.x] = oi;
        }
      }
      __syncthreads();
    }
    if (threadIdx.x == 0) {
      const int w = ri[0];
      idxOut[(size_t)b * S + j] = w;
      lastp[0] = Pb[w*3+0]; lastp[1] = Pb[w*3+1]; lastp[2] = Pb[w*3+2];
    }
    __syncthreads();
  }
}

// ---------------------------------------------------------------------------
// Dilated kNN: one block per query. kd argmin extractions (ascending, stable),
// keep every dil-th index. Matches top_k(-d2, k*d)[..., ::d].
// ---------------------------------------------------------------------------
__global__ __launch_bounds__(256) void knn_kernel(
    const float* __restrict__ Q, const float* __restrict__ P,
    int S, int Np, int kd, int dil, int k, int* __restrict__ idxOut)
{
  extern __shared__ float dist[];             // Np floats
  __shared__ float rv[256];
  __shared__ int ri[256];
  const int b = blockIdx.x / S, s = blockIdx.x % S;
  const float qx = Q[((size_t)b*S + s)*3+0];
  const float qy = Q[((size_t)b*S + s)*3+1];
  const float qz = Q[((size_t)b*S + s)*3+2];
  const float* Pb = P + (size_t)b * Np * 3;
  for (int i = threadIdx.x; i < Np; i += 256) {
    const float dx = Pb[i*3+0]-qx, dy = Pb[i*3+1]-qy, dz = Pb[i*3+2]-qz;
    dist[i] = dx*dx + dy*dy + dz*dz;
  }
  __syncthreads();
  for (int j = 0; j < kd; ++j) {
    float best = 3.4e38f; int bi = 0x7fffffff;
    for (int i = threadIdx.x; i < Np; i += 256) {
      const float v = dist[i];
      if (v < best || (v == best && i < bi)) { best = v; bi = i; }
    }
    rv[threadIdx.x] = best; ri[threadIdx.x] = bi;
    __syncthreads();
    for (int st = 128; st > 0; st >>= 1) {
      if (threadIdx.x < st) {
        const float ov = rv[threadIdx.x + st]; const int oi = ri[threadIdx.x + st];
        if (ov < rv[threadIdx.x] || (ov == rv[threadIdx.x] && oi < ri[threadIdx.x])) {
          rv[threadIdx.x] = ov; ri[threadIdx.x] = oi;
        }
      }
      __syncthreads();
    }
    if (threadIdx.x == 0) {
      const int w = ri[0];
      dist[w] = 3.4e38f;
      if (j % dil == 0) idxOut[((size_t)b*S + s) * k + j / dil] = w;
    }
    __syncthreads();
  }
}

// q[b,s,:] = p[b, idx[b,s], :]
__global__ void gather_points_kernel(const float* __restrict__ P,
                                     const int* __restrict__ idx,
                                     float* __restrict__ Q,
                                     int Np, int S, int total)
{
  const int t = blockIdx.x * 256 + threadIdx.x;
  if (t >= total) return;
  const int b = t / S;
  const int i = idx[t];
  const float* src = P + ((size_t)b * Np + i) * 3;
  Q[(size_t)t*3+0] = src[0]; Q[(size_t)t*3+1] = src[1]; Q[(size_t)t*3+2] = src[2];
}

// local[b,s,j,:] = p[b, idx[b,s,j], :] - q[b,s,:]
__global__ void local_coords_kernel(const float* __restrict__ P,
                                    const float* __restrict__ Q,
                                    const int* __restrict__ idx,
                                    float* __restrict__ L,
                                    int Np, int S, int k, int total)
{
  const int t = blockIdx.x * 256 + threadIdx.x;
  if (t >= total) return;
  const int bs = t / k;
  const int b = bs / S;
  const int i = idx[t];
  const float* pp = P + ((size_t)b * Np + i) * 3;
  const float* qq = Q + (size_t)bs * 3;
  L[(size_t)t*3+0] = pp[0] - qq[0];
  L[(size_t)t*3+1] = pp[1] - qq[1];
  L[(size_t)t*3+2] = pp[2] - qq[2];
}

// feats[row, 0:cin] = xf[b, idx[row], :] ; feats[row, cin:] = f[row, :]
__global__ void concat_feats_kernel(const float* __restrict__ Xf,
                                    const float* __restrict__ F,
                                    const int* __restrict__ idx,
                                    float* __restrict__ out,
                                    int Np, int S, int k, int cin, int mid,
                                    long total)
{
  const long t = (long)blockIdx.x * 256 + threadIdx.x;
  if (t >= total) return;
  const int C = cin + mid;
  const long row = t / C;
  const int c = (int)(t % C);
  const int b = (int)(row / ((long)S * k));
  if (c < cin) {
    const int i = idx[row];
    out[t] = Xf[((size_t)b * Np + i) * cin + c];
  } else {
    out[t] = F[row * (long)mid + (c - cin)];
  }
}

// Training-mode BN statistics: per channel over R rows.
__global__ __launch_bounds__(256) void bn_stats_kernel(
    const float* __restrict__ X, float* __restrict__ mean,
    float* __restrict__ var, int R, int C)
{
  const int c = blockIdx.x;
  __shared__ float ssum[256], ssq[256];
  float s = 0.0f, q = 0.0f;
  for (int r = threadIdx.x; r < R; r += 256) {
    const float v = X[(size_t)r * C + c];
    s += v; q += v * v;
  }
  ssum[threadIdx.x] = s; ssq[threadIdx.x] = q;
  __syncthreads();
  for (int st = 128; st > 0; st >>= 1) {
    if (threadIdx.x < st) {
      ssum[threadIdx.x] += ssum[threadIdx.x + st];
      ssq[threadIdx.x]  += ssq[threadIdx.x + st];
    }
    __syncthreads();
  }
  if (threadIdx.x == 0) {
    const float m = ssum[0] / (float)R;
    mean[c] = m;
    var[c] = ssq[0] / (float)R - m * m;
  }
}

// y = (x-m)*rsqrt(v+eps)*g + b ; optional SELU. In-place.
__global__ void bn_apply_kernel(float* __restrict__ X,
                                const float* __restrict__ mean,
                                const float* __restrict__ var,
                                const float* __restrict__ g,
                                const float* __restrict__ b,
                                long n, int C, int doSelu)
{
  const long t = (long)blockIdx.x * 256 + threadIdx.x;
  if (t >= n) return;
  const int c = (int)(t % C);
  float y = (X[t] - mean[c]) * rsqrtf(var[c] + BN_EPS) * g[c] + b[c];
  if (doSelu) {
    y = SELU_SCALE * (y > 0.0f ? y : SELU_ALPHA * (__expf(y) - 1.0f));
  }
  X[t] = y;
}

// out[b, c, s] = in[b, s, c]
__global__ void transpose_out_kernel(const float* __restrict__ in,
                                     float* __restrict__ out,
                                     int S, int C, long total)
{
  const long t = (long)blockIdx.x * 256 + threadIdx.x;
  if (t >= total) return;
  const long cs = (long)C * S;
  const int b = (int)(t / cs);
  const long rem = t % cs;
  const int c = (int)(rem / S);
  const int s = (int)(rem % S);
  out[t] = in[((size_t)b * S + s) * C + c];
}

// ---------------------------------------------------------------------------
// Host orchestration
// ---------------------------------------------------------------------------
namespace {
struct Lin { const float* w; const float* b; };
struct Bnp { const float* g; const float* b; };
struct XP {
  Lin lift1; Bnp bn1; Lin lift2; Bnp bn2;
  Lin t1; Bnp bnt1; Lin t2; Bnp bnt2; Lin t3; Lin out;
};
}

extern "C" void kernel_launch(void* const* d_in, const int* in_sizes, int n_in,
                              void* d_out, int out_size, void* d_ws, size_t ws_size,
                              hipStream_t stream) {
  (void)in_sizes; (void)n_in; (void)out_size; (void)ws_size;
  constexpr int Bz = 4, Npts = 8192;

  const float* P0 = (const float*)d_in[0];
  const float* X0 = (const float*)d_in[1];
  int ii = 2;
  auto nl = [&]() { Lin l{(const float*)d_in[ii], (const float*)d_in[ii + 1]}; ii += 2; return l; };
  auto nb = [&]() { Bnp b{(const float*)d_in[ii], (const float*)d_in[ii + 1]}; ii += 2; return b; };
  auto nx = [&]() {
    XP xp;
    xp.lift1 = nl(); xp.bn1 = nb();
    xp.lift2 = nl(); xp.bn2 = nb();
    xp.t1 = nl(); xp.bnt1 = nb();
    xp.t2 = nl(); xp.bnt2 = nb();
    xp.t3 = nl(); xp.out = nl();
    return xp;
  };
  XP c1 = nx(), c2 = nx(), c3 = nx(), c4 = nx(), dd3 = nx(), dd2 = nx(), dd1 = nx();
  Lin m1 = nl(); Bnp mbn1 = nb();
  Lin m2 = nl(); Bnp mbn2 = nb();
  Lin fc = nl();

  // Bump allocator over d_ws
  char* wsBase = (char*)d_ws;
  size_t wsOff = 0;
  auto alloc = [&](size_t bytes) -> void* {
    void* r = wsBase + wsOff;
    wsOff += (bytes + 255) & ~size_t(255);
    return r;
  };

  // Persistent buffers
  float* statMean = (float*)alloc(1024 * 4);
  float* statVar  = (float*)alloc(1024 * 4);
  int* fidx1 = (int*)alloc((size_t)Bz * 2048 * 4);
  int* fidx2 = (int*)alloc((size_t)Bz * 768 * 4);
  int* fidx3 = (int*)alloc((size_t)Bz * 384 * 4);
  int* fidx4 = (int*)alloc((size_t)Bz * 128 * 4);
  float* q1 = (float*)alloc((size_t)Bz * 2048 * 3 * 4);
  float* q2 = (float*)alloc((size_t)Bz * 768 * 3 * 4);
  float* q3 = (float*)alloc((size_t)Bz * 384 * 3 * 4);
  float* q4 = (float*)alloc((size_t)Bz * 128 * 3 * 4);
  float* x1 = (float*)alloc((size_t)Bz * 2048 * 256 * 4);
  float* x2 = (float*)alloc((size_t)Bz * 768 * 256 * 4);
  float* x3 = (float*)alloc((size_t)Bz * 384 * 512 * 4);
  float* x4 = (float*)alloc((size_t)Bz * 128 * 1024 * 4);
  float* y3 = (float*)alloc((size_t)Bz * 384 * 512 * 4);
  float* y2 = (float*)alloc((size_t)Bz * 768 * 256 * 4);
  float* y1 = (float*)alloc((size_t)Bz * 2048 * 256 * 4);
  float* h1 = (float*)alloc((size_t)Bz * 2048 * 256 * 4);
  float* h2 = (float*)alloc((size_t)Bz * 2048 * 128 * 4);
  float* logits = (float*)alloc((size_t)Bz * 2048 * 50 * 4);

  auto gemm = [&](const float* A, const Lin& lin, float* Cout, long M, int N, int K) {
    dim3 grid((N + 31) / 32, (unsigned)((M + 63) / 64));
    wmma_gemm_f32<<<grid, 256, 0, stream>>>(A, lin.w, lin.b, Cout, (int)M, N, K);
  };
  auto bn_selu = [&](float* buf, const Bnp& bnp, long R, int C, bool selu) {
    bn_stats_kernel<<<C, 256, 0, stream>>>(buf, statMean, statVar, (int)R, C);
    const long n = R * (long)C;
    bn_apply_kernel<<<(unsigned)((n + 255) / 256), 256, 0, stream>>>(
        buf, statMean, statVar, bnp.g, bnp.b, n, C, selu ? 1 : 0);
  };

  auto xconv = [&](const XP& pp, const float* Psup, int Np, const float* Xf, int cin,
                   const float* Q, int S, int k, int dil, int cout, float* OUT) {
    const size_t mark = wsOff;
    const int mid = cout / 4, kk2 = k * k, Cc = cin + mid;
    const long BS = (long)Bz * S, BSk = BS * k;

    int* idx = (int*)alloc((size_t)BSk * 4);
    float* local = (float*)alloc((size_t)BSk * 3 * 4);
    float* fA = (float*)alloc((size_t)BSk * mid * 4);
    float* fB = (float*)alloc((size_t)BSk * mid * 4);
    float* Xa = (float*)alloc((size_t)BS * kk2 * 4);
    float* Xb = (float*)alloc((size_t)BS * kk2 * 4);
    float* feats = (float*)alloc((size_t)BSk * Cc * 4);
    float* gbuf = (float*)alloc((size_t)BSk * Cc * 4);

    // 1) dilated kNN
    knn_kernel<<<Bz * S, 256, Np * sizeof(float), stream>>>(
        Q, Psup, S, Np, k * dil, dil, k, idx);
    // 2) local coords
    local_coords_kernel<<<(unsigned)((BSk + 255) / 256), 256, 0, stream>>>(
        Psup, Q, idx, local, Np, S, k, (int)BSk);
    // 3-4) lift MLP
    gemm(local, pp.lift1, fA, BSk, mid, 3);
    bn_selu(fA, pp.bn1, BSk, mid, true);
    gemm(fA, pp.lift2, fB, BSk, mid, mid);
    bn_selu(fB, pp.bn2, BSk, mid, true);
    // 5-8) X-transform MLP ([BS, 3k] view of local is contiguous)
    gemm(local, pp.t1, Xa, BS, kk2, 3 * k);
    bn_selu(Xa, pp.bnt1, BS, kk2, true);
    gemm(Xa, pp.t2, Xb, BS, kk2, kk2);
    bn_selu(Xb, pp.bnt2, BS, kk2, true);
    gemm(Xb, pp.t3, Xa, BS, kk2, kk2);   // X in Xa
    // 9) feats = concat(gather(x, idx), f)
    {
      const long total = BSk * Cc;
      concat_feats_kernel<<<(unsigned)((total + 255) / 256), 256, 0, stream>>>(
          Xf, fB, idx, feats, Np, S, k, cin, mid, total);
    }
    // 10) g = X @ feats (per (b,s))
    {
      const long waves = BS * ((Cc + 15) / 16);
      wmma_bmm_xfeats<<<(unsigned)((waves + 7) / 8), 256, 0, stream>>>(
          Xa, feats, gbuf, (int)BS, k, Cc);
    }
    // 11) out projection
    gemm(gbuf, pp.out, OUT, BS, cout, k * Cc);

    wsOff = mark;
  };

  // ---- FPS chain + query gathers ----
  fps_kernel<<<Bz, 512, Npts * sizeof(float), stream>>>(P0, Npts, 2048, fidx1);
  gather_points_kernel<<<(Bz * 2048 + 255) / 256, 256, 0, stream>>>(
      P0, fidx1, q1, Npts, 2048, Bz * 2048);
  fps_kernel<<<Bz, 512, 2048 * sizeof(float), stream>>>(q1, 2048, 768, fidx2);
  gather_points_kernel<<<(Bz * 768 + 255) / 256, 256, 0, stream>>>(
      q1, fidx2, q2, 2048, 768, Bz * 768);
  fps_kernel<<<Bz, 512, 768 * sizeof(float), stream>>>(q2, 768, 384, fidx3);
  gather_points_kernel<<<(Bz * 384 + 255) / 256, 256, 0, stream>>>(
      q2, fidx3, q3, 768, 384, Bz * 384);
  fps_kernel<<<Bz, 512, 384 * sizeof(float), stream>>>(q3, 384, 128, fidx4);
  gather_points_kernel<<<(Bz * 128 + 255) / 256, 256, 0, stream>>>(
      q3, fidx4, q4, 384, 128, Bz * 128);

  // ---- Encoder ----
  xconv(c1, P0, Npts, X0, 3,   q1, 2048, 8,  1, 256,  x1);
  xconv(c2, q1, 2048, x1, 256, q2, 768,  12, 2, 256,  x2);
  xconv(c3, q2, 768,  x2, 256, q3, 384,  16, 2, 512,  x3);
  xconv(c4, q3, 384,  x3, 512, q4, 128,  16, 6, 1024, x4);
  // ---- Decoder ----
  xconv(dd3, q4, 128, x4, 1024, q3, 384,  16, 6, 512, y3);
  xconv(dd2, q3, 384, y3, 512,  q2, 768,  12, 4, 256, y2);
  xconv(dd1, q2, 768, y2, 256,  q1, 2048, 8,  4, 256, y1);

  // ---- Head MLP ----
  const long Mh = (long)Bz * 2048;
  gemm(y1, m1, h1, Mh, 256, 256);
  bn_selu(h1, mbn1, Mh, 256, true);
  gemm(h1, m2, h2, Mh, 128, 256);
  bn_selu(h2, mbn2, Mh, 128, true);
  gemm(h2, fc, logits, Mh, 50, 128);

  // ---- [B, S, 50] -> [B, 50, S] ----
  const long total = (long)Bz * 50 * 2048;
  transpose_out_kernel<<<(unsigned)((total + 255) / 256), 256, 0, stream>>>(
      logits, (float*)d_out, 2048, 50, total);
}